// testLayer_395136991426
// MI455X (gfx1250) — compile-verified
//
#include <hip/hip_runtime.h>

// AdEx neuron simulation on MI455X (gfx1250).
// Memory-bound streaming scan: I[T,N] read once, v_trace[T,N] written once,
// neuron state (v, c, ref) in registers; one thread per neuron (max wave count).
// Roofline: 400 MB @ 23.3 TB/s => ~17 us floor. No matrix structure -> no WMMA;
// CDNA5 levers used: NT cache hints (touch-once streams > 192 MB L2),
// global_prefetch_b8, v_exp_f32, SALU lane-mask logic, wave32 occupancy.

#define ADEX_REF_STEPS 2

struct AdexState {
    float v;
    float c;
    int   ref;
};

// One forward-Euler AdEx step; returns v_out (also new s.v).
// ~18 VALU + 1 transcendental per call; divergent bools become lane masks
// combined on SALU.
__device__ __forceinline__ float adex_step(AdexState& s, float I_t)
{
    const float EL      = -70.6e-3f;
    const float VT      = -50.4e-3f;
    const float GL      = 30e-9f;
    const float GLDT    = 30e-9f * 2e-3f;              // GL * DELTAT
    const float DT_CM   = 1e-3f / 281e-12f;            // DT / CM
    const float DT_TAUW = 1e-3f / 144e-3f;             // DT / TAUW
    const float Aconst  = 4e-9f;
    const float Bconst  = 0.0805e-9f;
    // exp(x) = exp2(x*log2e): fold log2e into 1/DELTAT; rescale clamp point.
    const float K1      = 1.4426950408889634f / 2e-3f; // log2e / DELTAT
    const float K2      = 50.4e-3f * (1.4426950408889634f / 2e-3f); // -VT*K1
    const float ARGMAX  = 15.0f * 1.4426950408889634f;

    const bool  in_ref = s.ref > 0;                    // v_cmp -> mask
    const float v_eff  = in_ref ? EL : s.v;            // cndmask
    const float t0     = v_eff - EL;

    // Single hardware transcendental; arg = (v_eff - VT)*K1 as one FMA.
    const float arg = fminf(__builtin_fmaf(v_eff, K1, K2), ARGMAX);
    const float e   = __builtin_amdgcn_exp2f(arg);

    // raw = v_eff + DT_CM * (-GL*t0 + GLDT*e - c + I)  (FMA chain)
    float acc = __builtin_fmaf(GLDT, e, I_t - s.c);
    acc       = __builtin_fmaf(-GL, t0, acc);
    const float raw = __builtin_fmaf(DT_CM, acc, v_eff);

    // v_new = in_ref ? EL : raw; EL < VT so in_ref lanes can never spike.
    //   spike = (raw >= VT) && !in_ref        (SALU mask andn2)
    //   v_out = (in_ref || raw >= VT) ? EL : raw  (single cndmask)
    const bool ge    = raw >= VT;                      // v_cmp -> mask
    const bool spike = ge && !in_ref;                  // SALU
    const bool fire  = in_ref || ge;                   // SALU
    const float v_out = fire ? EL : raw;               // cndmask

    // c_new = c + DT_TAUW * (A*t0 - c)  (two FMAs)
    const float c_new = __builtin_fmaf(DT_TAUW, __builtin_fmaf(Aconst, t0, -s.c), s.c);

    s.c   = spike ? (c_new + Bconst) : c_new;          // add + cndmask
    // max(ref-1, 0) == (in_ref ? ref-1 : 0) since ref >= 0; reuse in_ref mask.
    s.ref = spike ? ADEX_REF_STEPS : (in_ref ? s.ref - 1 : 0);
    s.v   = v_out;
    return v_out;
}

__global__ __launch_bounds__(256) void adex_kernel(
    const float* __restrict__ I,
    const float* __restrict__ v0,
    const float* __restrict__ c0,
    const int*   __restrict__ ref0,
    float*       __restrict__ out,
    int N, int T)
{
    const int i = blockIdx.x * blockDim.x + threadIdx.x;
    if (i >= N) return;

    __builtin_assume(T >= 2);
    __builtin_assume(N > 0);

    AdexState s;
    s.v   = v0[i];
    s.c   = c0[i];
    s.ref = ref0[i];

    // Uniform row offset + loop-invariant lane index: uniform part strength-
    // reduces to one SALU/VALU base update per unrolled body, rows t+1..t+3
    // addressed via 24-bit immediate offsets.
    float Icur = __builtin_nontemporal_load(I + i);

    const int Tm1 = T - 1;
    int t = 0;
    #pragma unroll 4
    for (; t < Tm1; ++t) {
        // Next row's current is independent of the recurrence: one-ahead load.
        float Inext = __builtin_nontemporal_load(I + (size_t)(t + 1) * (size_t)N + i);
        // Stream prefetch 8 rows (8*N*4 B) ahead of the demand-load front.
        const int tp = (t + 8 < T) ? (t + 8) : Tm1;
        __builtin_prefetch(I + (size_t)tp * (size_t)N + i, 0, 1);

        const float v_out = adex_step(s, Icur);
        __builtin_nontemporal_store(v_out, out + (size_t)t * (size_t)N + i);

        Icur = Inext;
    }

    // Peeled final step: hot loop above has no end-of-range clamps on the load.
    const float v_out = adex_step(s, Icur);
    __builtin_nontemporal_store(v_out, out + (size_t)Tm1 * (size_t)N + i);
}

extern "C" void kernel_launch(void* const* d_in, const int* in_sizes, int n_in,
                              void* d_out, int out_size, void* d_ws, size_t ws_size,
                              hipStream_t stream) {
    const float* I    = (const float*)d_in[0];   // [T, N]
    const float* v0   = (const float*)d_in[1];   // [N]
    const float* c0   = (const float*)d_in[2];   // [N]
    const int*   ref0 = (const int*)  d_in[3];   // [N]
    float*       out  = (float*)d_out;           // [T, N]

    const int N = in_sizes[1];
    const int T = in_sizes[0] / N;

    const int block = 256;                        // 8 wave32s per workgroup
    const int grid  = (N + block - 1) / block;    // ~3128 resident wave32s

    adex_kernel<<<grid, block, 0, stream>>>(I, v0, c0, ref0, out, N, T);
}